// AGCNBlock_32693291057970
// MI455X (gfx1250) — compile-verified
//
#include <hip/hip_runtime.h>
#include <hip/hip_bf16.h>

// Problem dims (fixed by the reference)
#define BB 8
#define NN 2048
#define DD 128
#define HH 128
#define KK 1434
#define KP 1536   // K padded to a multiple of 128 so all GEMM dims tile cleanly

typedef __attribute__((ext_vector_type(2))) float v2f;
typedef __attribute__((ext_vector_type(8))) float v8f;

__device__ __forceinline__ v8f wmma4(v2f a, v2f b, v8f c) {
    // D = A(16x4,f32) * B(4x16,f32) + C(16x16,f32)
    return __builtin_amdgcn_wmma_f32_16x16x4_f32(
        /*neg_a=*/false, a, /*neg_b=*/false, b,
        /*c_mod=*/(short)0, c, /*reuse_a=*/false, /*reuse_b=*/false);
}

// ---------------------------------------------------------------------------
// Generic batched f32 WMMA GEMM: C[b] = op(A[b] @ B[b])
//   block = 256 threads (8 waves, 4x2), block tile 128x128, wave tile 32x64
//   grid  = (N2/128, M/128, B);  M,N2 multiples of 128; Kc multiple of 4
//   TRANSB: B stored as [N2 rows x Kc] row-major (use row j as column j)
//   bias[col] + relu epilogue; rowscale[b*M+row] multiplier; guarded stores.
// ---------------------------------------------------------------------------
template <int TRANSB>
__global__ __launch_bounds__(256)
void gemm_wmma_f32(const float* __restrict__ A, const float* __restrict__ B,
                   float* __restrict__ C,
                   int M, int N2, int Kc,
                   long long sA, long long sB, long long sC,
                   int lda, int ldb, int ldc,
                   int M_store, int N_store,
                   const float* __restrict__ bias, int relu,
                   const float* __restrict__ rowscale)
{
    const int bz = blockIdx.z;
    A += (long long)bz * sA;
    B += (long long)bz * sB;
    C += (long long)bz * sC;

    const int tid  = threadIdx.x;
    const int wave = tid >> 5;
    const int lane = tid & 31;
    const int l    = lane & 15;   // position within half-wave
    const int lh   = lane >> 4;   // half-wave select
    const int wr   = wave >> 1;   // 4 wave-rows
    const int wc   = wave & 1;    // 2 wave-cols

    const int m0 = blockIdx.y * 128 + wr * 32;
    const int n0 = blockIdx.x * 128 + wc * 64;

    v8f acc[2][4];
    #pragma unroll
    for (int mi = 0; mi < 2; ++mi)
        #pragma unroll
        for (int ni = 0; ni < 4; ++ni) acc[mi][ni] = (v8f){};

    const float* Arow0 = A + (long long)(m0 + l)      * lda;  // rows m0..m0+15
    const float* Arow1 = A + (long long)(m0 + 16 + l) * lda;  // rows m0+16..m0+31

    for (int k = 0; k < Kc; k += 4) {
        const int kk = k + 2 * lh;
        // A frags: lane<16 -> {K=k,K=k+1}; lane>=16 -> {K=k+2,K=k+3}
        v2f a0 = *(const v2f*)(Arow0 + kk);
        v2f a1 = *(const v2f*)(Arow1 + kk);
        v2f bf[4];
        if (TRANSB) {
            #pragma unroll
            for (int ni = 0; ni < 4; ++ni)
                bf[ni] = *(const v2f*)(B + (long long)(n0 + 16 * ni + l) * ldb + kk);
        } else {
            const float* Bp = B + (long long)kk * ldb;
            #pragma unroll
            for (int ni = 0; ni < 4; ++ni) {
                bf[ni].x = Bp[n0 + 16 * ni + l];
                bf[ni].y = Bp[ldb + n0 + 16 * ni + l];
            }
        }
        // prefetch upcoming A tiles (speculative; OOB prefetch is dropped)
        __builtin_prefetch(Arow0 + kk + 64, 0, 1);
        __builtin_prefetch(Arow1 + kk + 64, 0, 1);

        #pragma unroll
        for (int ni = 0; ni < 4; ++ni) {
            acc[0][ni] = wmma4(a0, bf[ni], acc[0][ni]);
            acc[1][ni] = wmma4(a1, bf[ni], acc[1][ni]);
        }
    }

    // Store: VGPR i of a 16x16 frag -> row base+i (lanes<16) / base+i+8 (lanes>=16)
    #pragma unroll
    for (int mi = 0; mi < 2; ++mi) {
        #pragma unroll
        for (int ni = 0; ni < 4; ++ni) {
            const int col = n0 + 16 * ni + l;
            if (col >= N_store) continue;
            const float bval = bias ? bias[col] : 0.0f;
            const int row_base = m0 + 16 * mi + 8 * lh;
            #pragma unroll
            for (int i = 0; i < 8; ++i) {
                const int row = row_base + i;
                if (row < M_store) {
                    float v = acc[mi][ni][i] + bval;
                    if (relu) v = fmaxf(v, 0.0f);
                    if (rowscale) v *= rowscale[(long long)bz * M + row];
                    C[(long long)row * ldc + col] = v;
                }
            }
        }
    }
}

// ---------------------------------------------------------------------------
// Attention / pooling helper kernels
// ---------------------------------------------------------------------------

// att_raw[b,n] = hidden[b,n,:] . w_b  + (mask-1)*BIG
__global__ void k_dot(const float* __restrict__ hidden, const float* __restrict__ wb,
                      const float* __restrict__ mask, float* __restrict__ attraw)
{
    int t = blockIdx.x * blockDim.x + threadIdx.x;
    if (t >= BB * NN) return;
    const float* hr = hidden + (long long)t * HH;
    float s = 0.0f;
    #pragma unroll 8
    for (int h = 0; h < HH; ++h) s += hr[h] * wb[h];
    attraw[t] = s + (mask[t] - 1.0f) * 1e10f;
}

__global__ void k_rowmax(const float* __restrict__ attraw, float* __restrict__ maxv)
{
    int b = blockIdx.x, tid = threadIdx.x;
    __shared__ float sm[256];
    float m = -3.4e38f;
    for (int n = tid; n < NN; n += 256) m = fmaxf(m, attraw[b * NN + n]);
    sm[tid] = m; __syncthreads();
    for (int s = 128; s > 0; s >>= 1) {
        if (tid < s) sm[tid] = fmaxf(sm[tid], sm[tid + s]);
        __syncthreads();
    }
    if (tid == 0) maxv[b] = sm[0];
}

__global__ void k_exp(const float* __restrict__ attraw, const float* __restrict__ maxv,
                      const float* __restrict__ tau, float* __restrict__ e)
{
    int t = blockIdx.x * blockDim.x + threadIdx.x;
    if (t >= BB * NN) return;
    int b = t / NN;
    e[t] = expf((attraw[t] - maxv[b]) * fabsf(*tau));
}

// denom[b,n] = sum_m adj[b,n,m]*e[b,m] + EPS ; diag[b,n] = adj[b,n,n]
__global__ void k_denomdiag(const float* __restrict__ adj, const float* __restrict__ e,
                            float* __restrict__ denom, float* __restrict__ diag)
{
    int n = blockIdx.x, b = blockIdx.y, tid = threadIdx.x;
    const float* row = adj + ((long long)b * NN + n) * NN;
    const float* eb  = e + b * NN;
    float s = 0.0f;
    for (int m = tid; m < NN; m += 256) s += row[m] * eb[m];
    __shared__ float sm[256];
    sm[tid] = s; __syncthreads();
    for (int st = 128; st > 0; st >>= 1) {
        if (tid < st) sm[tid] += sm[tid + st];
        __syncthreads();
    }
    if (tid == 0) {
        denom[b * NN + n] = sm[0] + 1e-10f;
        diag[b * NN + n]  = row[n];
    }
}

// colsum[b,m] = sum_n adj[b,n,m]
__global__ void k_colsum(const float* __restrict__ adj, float* __restrict__ colsum)
{
    int t = blockIdx.x * blockDim.x + threadIdx.x;
    if (t >= BB * NN) return;
    int b = t / NN, m = t % NN;
    const float* base = adj + (long long)b * NN * NN + m;
    float s = 0.0f;
    for (int n = 0; n < NN; ++n) s += base[(long long)n * NN];
    colsum[t] = s;
}

// Z[b,n,h] = att[b,n] * hidden[b,n,h]
__global__ void k_attz(const float* __restrict__ e, const float* __restrict__ diag,
                       const float* __restrict__ denom, const float* __restrict__ colsum,
                       const float* __restrict__ mask, const float* __restrict__ hidden,
                       float* __restrict__ Z)
{
    long long t = (long long)blockIdx.x * blockDim.x + threadIdx.x;
    if (t >= (long long)BB * NN * HH) return;
    long long bn = t / HH;
    float att = e[bn] * diag[bn] / denom[bn] * colsum[bn] * mask[bn];
    Z[t] = att * hidden[t];
}

// out[b,h] = sum_n mask*hidden / sum_n mask
__global__ void k_meanpool(const float* __restrict__ hidden, const float* __restrict__ mask,
                           float* __restrict__ out)
{
    int t = blockIdx.x * blockDim.x + threadIdx.x;
    if (t >= BB * HH) return;
    int b = t / HH, h = t % HH;
    const float* hb = hidden + (long long)b * NN * HH + h;
    const float* mb = mask + b * NN;
    float s = 0.0f, sm = 0.0f;
    for (int n = 0; n < NN; ++n) { s += mb[n] * hb[(long long)n * HH]; sm += mb[n]; }
    out[t] = s / sm;
}

// S[b,n] = sum_k adj[b, top[b,k], n]
__global__ void k_S(const float* __restrict__ adj, const int* __restrict__ top,
                    float* __restrict__ S)
{
    int t = blockIdx.x * blockDim.x + threadIdx.x;
    if (t >= BB * NN) return;
    int b = t / NN, n = t % NN;
    const int*   tp = top + b * KK;
    const float* ab = adj + (long long)b * NN * NN;
    float s = 0.0f;
    for (int k = 0; k < KK; ++k) s += ab[(long long)tp[k] * NN + n];
    S[t] = s;
}

// assign_m[b,kp,n] = kp<KK ? adj[b,top[b,kp],n]/(S[b,n]+EPS) : 0   (zero-padded rows)
__global__ void k_assign(const float* __restrict__ adj, const int* __restrict__ top,
                         const float* __restrict__ S, float* __restrict__ assign)
{
    long long t = (long long)blockIdx.x * blockDim.x + threadIdx.x;
    if (t >= (long long)BB * KP * NN) return;
    int n = (int)(t % NN);
    long long r = t / NN;
    int kp = (int)(r % KP);
    int b  = (int)(r / KP);
    float v = 0.0f;
    if (kp < KK) {
        int tr = top[b * KK + kp];
        v = adj[(long long)b * NN * NN + (long long)tr * NN + n] / (S[b * NN + n] + 1e-10f);
    }
    assign[t] = v;
}

__global__ void k_newmask(float* __restrict__ out3)
{
    int t = blockIdx.x * blockDim.x + threadIdx.x;
    if (t < BB * KK) out3[t] = 1.0f;
}

// ---------------------------------------------------------------------------
extern "C" void kernel_launch(void* const* d_in, const int* in_sizes, int n_in,
                              void* d_out, int out_size, void* d_ws, size_t ws_size,
                              hipStream_t stream)
{
    const float* X    = (const float*)d_in[0];  // [8,2048,128]
    const float* adj  = (const float*)d_in[1];  // [8,2048,2048]
    const float* mask = (const float*)d_in[2];  // [8,2048]
    const float* W1   = (const float*)d_in[3];  // [128,128]
    const float* b1   = (const float*)d_in[4];  // [128]
    const float* W2   = (const float*)d_in[5];  // [128,128]
    const float* b2   = (const float*)d_in[6];  // [128]
    const float* wb   = (const float*)d_in[7];  // [1,128,1]
    const float* tau  = (const float*)d_in[8];  // [1]
    const int*   top  = (const int*)  d_in[9];  // [8,1434]

    float* out = (float*)d_out;
    const long long OUT0 = 0;                                   // out      [8,128]
    const long long OUT1 = OUT0 + (long long)BB * HH;           // H_out    [8,1434,128]
    const long long OUT2 = OUT1 + (long long)BB * KK * HH;      // new_adj  [8,1434,1434]
    const long long OUT3 = OUT2 + (long long)BB * KK * KK;      // new_mask [8,1434]

    // Workspace carve (floats)
    float* ws = (float*)d_ws;
    const long long SZ_NH = (long long)BB * NN * HH;            // 2,097,152
    float* AX     = ws;                    // adj @ X
    float* h1     = AX     + SZ_NH;
    float* Ah     = h1     + SZ_NH;        // adj @ h1
    float* hidden = Ah     + SZ_NH;
    float* Z      = hidden + SZ_NH;
    float* attraw = Z      + SZ_NH;
    float* e      = attraw + (long long)BB * NN;
    float* denom  = e      + (long long)BB * NN;
    float* diag   = denom  + (long long)BB * NN;
    float* colsum = diag   + (long long)BB * NN;
    float* S      = colsum + (long long)BB * NN;
    float* maxv   = S      + (long long)BB * NN;
    float* assign = maxv   + 8;            // [8,1536,2048]
    float* T1     = assign + (long long)BB * KP * NN;  // assign @ adj, [8,1536,2048]

    const dim3 blk(256);

    // --- GCN layer 1: AX = adj @ X ; h1 = relu(AX @ W1 + b1) ---
    gemm_wmma_f32<0><<<dim3(DD / 128, NN / 128, BB), blk, 0, stream>>>(
        adj, X, AX, NN, DD, NN,
        (long long)NN * NN, (long long)NN * DD, (long long)NN * DD,
        NN, DD, DD, NN, DD, nullptr, 0, nullptr);
    gemm_wmma_f32<0><<<dim3(HH / 128, NN / 128, BB), blk, 0, stream>>>(
        AX, W1, h1, NN, HH, DD,
        (long long)NN * DD, 0LL, (long long)NN * HH,
        DD, HH, HH, NN, HH, b1, 1, nullptr);

    // --- GCN layer 2: Ah = adj @ h1 ; hidden = mask * relu(Ah @ W2 + b2) ---
    gemm_wmma_f32<0><<<dim3(HH / 128, NN / 128, BB), blk, 0, stream>>>(
        adj, h1, Ah, NN, HH, NN,
        (long long)NN * NN, (long long)NN * HH, (long long)NN * HH,
        NN, HH, HH, NN, HH, nullptr, 0, nullptr);
    gemm_wmma_f32<0><<<dim3(HH / 128, NN / 128, BB), blk, 0, stream>>>(
        Ah, W2, hidden, NN, HH, HH,
        (long long)NN * HH, 0LL, (long long)NN * HH,
        HH, HH, HH, NN, HH, b2, 1, mask);

    // --- neighbor-softmax attention ---
    k_dot      <<<(BB * NN + 255) / 256, blk, 0, stream>>>(hidden, wb, mask, attraw);
    k_rowmax   <<<BB, blk, 0, stream>>>(attraw, maxv);
    k_exp      <<<(BB * NN + 255) / 256, blk, 0, stream>>>(attraw, maxv, tau, e);
    k_denomdiag<<<dim3(NN, BB), blk, 0, stream>>>(adj, e, denom, diag);
    k_colsum   <<<(BB * NN + 255) / 256, blk, 0, stream>>>(adj, colsum);
    k_attz     <<<(int)(((long long)BB * NN * HH + 255) / 256), blk, 0, stream>>>(
        e, diag, denom, colsum, mask, hidden, Z);
    k_meanpool <<<(BB * HH + 255) / 256, blk, 0, stream>>>(hidden, mask, out + OUT0);

    // --- pooling: assign_m (normalized gathered rows, zero-padded to KP) ---
    k_S     <<<(BB * NN + 255) / 256, blk, 0, stream>>>(adj, top, S);
    k_assign<<<(int)(((long long)BB * KP * NN + 255) / 256), blk, 0, stream>>>(
        adj, top, S, assign);

    // H_out = assign_m @ Z  -> out[OUT1], [8,1434,128]
    gemm_wmma_f32<0><<<dim3(HH / 128, KP / 128, BB), blk, 0, stream>>>(
        assign, Z, out + OUT1, KP, HH, NN,
        (long long)KP * NN, (long long)NN * HH, (long long)KK * HH,
        NN, HH, HH, KK, HH, nullptr, 0, nullptr);

    // T1 = assign_m @ adj  [8,1536,2048]
    gemm_wmma_f32<0><<<dim3(NN / 128, KP / 128, BB), blk, 0, stream>>>(
        assign, adj, T1, KP, NN, NN,
        (long long)KP * NN, (long long)NN * NN, (long long)KP * NN,
        NN, NN, NN, KP, NN, nullptr, 0, nullptr);

    // new_adj = T1 @ assign_m^T -> out[OUT2], [8,1434,1434]
    gemm_wmma_f32<1><<<dim3(KP / 128, KP / 128, BB), blk, 0, stream>>>(
        T1, assign, out + OUT2, KP, KP, NN,
        (long long)KP * NN, (long long)KP * NN, (long long)KK * KK,
        NN, NN, KK, KK, KK, nullptr, 0, nullptr);

    k_newmask<<<(BB * KK + 255) / 256, blk, 0, stream>>>(out + OUT3);
}